// OutputLayer_41961830482215
// MI455X (gfx1250) — compile-verified
//
#include <hip/hip_runtime.h>
#include <hip/hip_bf16.h>

#define N_LEAVES 256
#define N_CLASSES 512
#define BATCH 131072

// Native clang vector type: accepted by __builtin_nontemporal_{load,store},
// lowers to 128-bit global_load_b128 / global_store_b128 with NT hint.
typedef float v4f __attribute__((ext_vector_type(4)));

// One wave32 per batch row.
//  - 32 lanes x 8 weights (2x v4f, NT loads): coalesced 1024B row read
//  - per-lane argmax in ascending-index order (first-occurrence semantics)
//  - 5-step XOR butterfly reduction (value desc, index asc on ties)
//  - scalar index via readfirstlane, then 32 lanes x 4 v4f gather from
//    opinions (L2-hot, regular temporal) and NT store of the output row.
__global__ __launch_bounds__(256) void argmax_gather_kernel(
    const float* __restrict__ opinions,   // [256, 512]
    const float* __restrict__ weights,    // [131072, 256]
    float* __restrict__ out)              // [131072, 512]
{
    const int lane = threadIdx.x & 31;
    const int wave = threadIdx.x >> 5;
    const int row  = blockIdx.x * 8 + wave;   // gridDim.x == BATCH/8

    // ---- load this row's 256 weights: 2 x v4f per lane, non-temporal ----
    const v4f* wrow =
        reinterpret_cast<const v4f*>(weights + (size_t)row * N_LEAVES);
    const v4f a = __builtin_nontemporal_load(&wrow[lane]);        // idx lane*4 + 0..3
    const v4f b = __builtin_nontemporal_load(&wrow[lane + 32]);   // idx 128 + lane*4 + 0..3

    float v[8]   = {a.x, a.y, a.z, a.w, b.x, b.y, b.z, b.w};
    const int i0 = lane * 4;
    int ids[8]   = {i0, i0 + 1, i0 + 2, i0 + 3,
                    128 + i0, 128 + i0 + 1, 128 + i0 + 2, 128 + i0 + 3};

    float best = v[0];
    int   bi   = ids[0];
#pragma unroll
    for (int j = 1; j < 8; ++j) {
        if (v[j] > best) { best = v[j]; bi = ids[j]; }   // strict >: keep first occurrence
    }

    // ---- wave32 butterfly argmax reduction ----
#pragma unroll
    for (int off = 16; off >= 1; off >>= 1) {
        const float ov = __shfl_xor(best, off, 32);
        const int   oi = __shfl_xor(bi,   off, 32);
        if (ov > best || (ov == best && oi < bi)) { best = ov; bi = oi; }
    }
    const int idx = __builtin_amdgcn_readfirstlane(bi);  // uniform across the wave

    // ---- gather the winning opinion row (L2-resident) and stream it out ----
    const v4f* src =
        reinterpret_cast<const v4f*>(opinions + (size_t)idx * N_CLASSES);
    v4f* dst = reinterpret_cast<v4f*>(out + (size_t)row * N_CLASSES);
#pragma unroll
    for (int j = 0; j < 4; ++j) {
        const v4f t = src[lane + 32 * j];                // regular temporal: keep hot in L2
        __builtin_nontemporal_store(t, &dst[lane + 32 * j]);  // NT: don't pollute L2
    }
}

extern "C" void kernel_launch(void* const* d_in, const int* in_sizes, int n_in,
                              void* d_out, int out_size, void* d_ws, size_t ws_size,
                              hipStream_t stream) {
    (void)in_sizes; (void)n_in; (void)d_ws; (void)ws_size; (void)out_size;
    const float* opinions = (const float*)d_in[0];  // [256, 512]
    const float* weights  = (const float*)d_in[1];  // [131072, 256]
    float* out            = (float*)d_out;          // [131072, 512]

    const int rows_per_block = 8;                   // 8 waves of 32
    dim3 grid(BATCH / rows_per_block);              // 16384 blocks
    dim3 block(256);
    argmax_gather_kernel<<<grid, block, 0, stream>>>(opinions, weights, out);
}